// DeformDownsampleBlock1_6313601925214
// MI455X (gfx1250) — compile-verified
//
#include <hip/hip_runtime.h>
#include <hip/hip_bf16.h>
#include <math.h>

typedef __attribute__((ext_vector_type(2))) float v2f;
typedef __attribute__((ext_vector_type(8))) float v8f;

#define CPG 8
#define BN_EPS 1e-5f

// ---------------------------------------------------------------------------
// CDNA5 async-copy + wait helpers (ASYNCcnt-tracked, ISA ch.10 / 15.18)
// ---------------------------------------------------------------------------
__device__ __forceinline__ void async_copy_b128(unsigned lds_addr, const void* gaddr) {
  asm volatile("global_load_async_to_lds_b128 %0, %1, off"
               :: "v"(lds_addr), "v"(gaddr) : "memory");
}
__device__ __forceinline__ void wait_async0() {
  asm volatile("s_wait_asynccnt 0" ::: "memory");
}

__device__ __forceinline__ v8f wmma_k4(v2f a, v2f b, v8f c) {
  return __builtin_amdgcn_wmma_f32_16x16x4_f32(false, a, false, b, (short)0, c,
                                               false, false);
}

// ---------------------------------------------------------------------------
// maxpool3d: kernel 3, stride 2, pad 1 (pad = -inf). x:[N,C,D,H,W] -> out:[N,C,Do,Ho,Wo]
// ---------------------------------------------------------------------------
__global__ void maxpool3d_kernel(const float* __restrict__ x, float* __restrict__ out,
                                 int N, int C, int D, int H, int W,
                                 int Do, int Ho, int Wo) {
  int S = Do * Ho * Wo;
  long total = (long)N * C * S;
  long idx = (long)blockIdx.x * blockDim.x + threadIdx.x;
  if (idx >= total) return;
  int s = (int)(idx % S);
  long nc = idx / S;
  int wo = s % Wo, ho = (s / Wo) % Ho, d0 = s / (Wo * Ho);
  const float* xp = x + nc * (long)D * H * W;
  float m = -INFINITY;
  for (int kd = 0; kd < 3; ++kd) {
    int d = d0 * 2 - 1 + kd; if (d < 0 || d >= D) continue;
    for (int kh = 0; kh < 3; ++kh) {
      int h = ho * 2 - 1 + kh; if (h < 0 || h >= H) continue;
      for (int kw = 0; kw < 3; ++kw) {
        int w = wo * 2 - 1 + kw; if (w < 0 || w >= W) continue;
        m = fmaxf(m, xp[((long)d * H + h) * W + w]);
      }
    }
  }
  out[idx] = m;
}

// ---------------------------------------------------------------------------
// im2col for dense 3x3x3 conv, pad 1. x:[N,C,D,H,W] channel-major.
// cols:[(n*S+s), K] with K = C*27, k = c*27 + t
// ---------------------------------------------------------------------------
__global__ void im2col3d_kernel(const float* __restrict__ x, float* __restrict__ cols,
                                int N, int C, int D, int H, int W,
                                int Do, int Ho, int Wo, int stride) {
  int S = Do * Ho * Wo;
  long total = (long)N * 27 * S;
  long idx = (long)blockIdx.x * blockDim.x + threadIdx.x;
  if (idx >= total) return;
  int s = (int)(idx % S);
  long tmp = idx / S;
  int t = (int)(tmp % 27);
  int n = (int)(tmp / 27);
  int wo = s % Wo, ho = (s / Wo) % Ho, d0 = s / (Wo * Ho);
  int ki = t / 9, kj = (t / 3) % 3, kk = t % 3;
  int d = d0 * stride - 1 + ki;
  int h = ho * stride - 1 + kj;
  int w = wo * stride - 1 + kk;
  bool ok = (d >= 0 && d < D && h >= 0 && h < H && w >= 0 && w < W);
  long SP = (long)D * H * W;
  long sin = ok ? ((long)d * H + h) * W + w : 0;
  int K = C * 27;
  float* cp = cols + ((long)n * S + s) * K + t;
  const float* xp = x + (long)n * C * SP + sin;
  for (int c = 0; c < C; ++c)
    cp[(long)c * 27] = ok ? xp[(long)c * SP] : 0.0f;
}

// ---------------------------------------------------------------------------
// Deformable sampling -> cols.  off:[(n*S+s), G*81] spatial-major,
// channel och = g*81 + coord*27 + t.  x:[N,C,D,H,W] channel-major.
// One thread per (n,g,t,s): trilinear corners computed once, gathered for
// all CPG channels of the group.
// ---------------------------------------------------------------------------
__global__ void deform_cols_kernel(const float* __restrict__ x, const float* __restrict__ off,
                                   float* __restrict__ cols,
                                   int N, int C, int D, int H, int W,
                                   int G, int Do, int Ho, int Wo, int stride) {
  int S = Do * Ho * Wo;
  long total = (long)N * G * 27 * S;
  long idx = (long)blockIdx.x * blockDim.x + threadIdx.x;
  if (idx >= total) return;
  int s = (int)(idx % S);
  long tmp = idx / S;
  int t = (int)(tmp % 27); tmp /= 27;
  int g = (int)(tmp % G);
  int n = (int)(tmp / G);
  int wo = s % Wo, ho = (s / Wo) % Ho, d0 = s / (Wo * Ho);
  int ki = t / 9, kj = (t / 3) % 3, kk = t % 3;
  int OC = G * 81;
  const float* op = off + ((long)n * S + s) * OC + g * 81 + t;
  float pd = op[0]  + (float)(d0 * stride - 1 + ki);
  float ph = op[27] + (float)(ho * stride - 1 + kj);
  float pw = op[54] + (float)(wo * stride - 1 + kk);
  float fd0 = floorf(pd), fh0 = floorf(ph), fw0 = floorf(pw);
  float fd = pd - fd0, fh = ph - fh0, fw = pw - fw0;
  int di0 = (int)fd0, hi0 = (int)fh0, wi0 = (int)fw0;
  float acc[CPG];
#pragma unroll
  for (int c = 0; c < CPG; ++c) acc[c] = 0.0f;
  long SP = (long)D * H * W;
  const float* xp = x + ((long)n * C + g * CPG) * SP;
#pragma unroll
  for (int cd = 0; cd < 2; ++cd) {
#pragma unroll
    for (int ch = 0; ch < 2; ++ch) {
#pragma unroll
      for (int cw = 0; cw < 2; ++cw) {
        int di = di0 + cd, hi = hi0 + ch, wi = wi0 + cw;
        float wt = (cd ? fd : 1.0f - fd) * (ch ? fh : 1.0f - fh) * (cw ? fw : 1.0f - fw);
        bool valid = (di >= 0 && di < D && hi >= 0 && hi < H && wi >= 0 && wi < W);
        if (valid && wt != 0.0f) {
          long ii = ((long)di * H + hi) * W + wi;
#pragma unroll
          for (int c = 0; c < CPG; ++c) acc[c] += wt * xp[(long)c * SP + ii];
        }
      }
    }
  }
  int K = C * 27;
  float* cp = cols + ((long)n * S + s) * K + (long)(g * CPG) * 27 + t;
#pragma unroll
  for (int c = 0; c < CPG; ++c) cp[(long)c * 27] = acc[c];
}

// ---------------------------------------------------------------------------
// f32 WMMA GEMM with double-buffered async-to-LDS staging.
//   Cmat[M,Ncol] = A[M,K] * Wt[Ncol,K]^T (+ bias)
// Block: 256 threads (8 waves, 4x2), block tile 128(M) x 64(N), K-chunk 16.
// Wave tile: 32x32 = 2x2 WMMA tiles -> 4 wmma per 4 fragment loads.
// LDS row stride padded to 20 floats (80B): 16B-aligned for b128 async
// writes, bank-conflict-free for the 16-row ds_load_b64 fragment reads.
// Requires: M % 128 == 0, K % 16 == 0 (true for all 4 GEMMs here).
// ---------------------------------------------------------------------------
#define LDSTRIDE 20
__global__ void gemm_wmma_f32_kernel(const float* __restrict__ A,
                                     const float* __restrict__ Wt,
                                     const float* __restrict__ bias,
                                     float* __restrict__ Cmat,
                                     int M, int K, int Ncol, int has_bias) {
  __shared__ float As[2][128 * LDSTRIDE];
  __shared__ float Bs[2][64 * LDSTRIDE];

  int tid = threadIdx.x;
  int lane = tid & 31;
  int wave = tid >> 5;
  int wm = wave >> 1;          // 0..3 -> M offset wm*32
  int wn = wave & 1;           // 0..1 -> N offset wn*32
  int half = lane >> 4;
  int row = lane & 15;
  long bm = (long)blockIdx.x * 128;
  int bn = blockIdx.y * 64;

  const float* Ag = A + bm * K;
  unsigned aBase = (unsigned)(size_t)(void*)&As[0][0];
  unsigned bBase = (unsigned)(size_t)(void*)&Bs[0][0];

  // B global row for this thread's tile chunk (clamped at the N edge; edge
  // columns are masked at the C store, so clamped data is never visible).
  int brow = tid >> 2;                  // 0..63
  int bq = tid & 3;                     // 16B chunk within the 16-float row
  int bgr = bn + brow; if (bgr >= Ncol) bgr = 0;

  auto load_tiles = [&](int buf, int kpos) {
    // A tile: 128 rows x 16 floats = 512 b128 chunks, 2 per thread
#pragma unroll
    for (int i = 0; i < 2; ++i) {
      int id = tid + i * 256;
      int r = id >> 2, q = id & 3;
      async_copy_b128(aBase + (unsigned)(buf * (128 * LDSTRIDE) + r * LDSTRIDE + q * 4) * 4u,
                      Ag + (long)r * K + kpos + q * 4);
    }
    // B tile: 64 rows x 16 floats = 256 b128 chunks, 1 per thread
    async_copy_b128(bBase + (unsigned)(buf * (64 * LDSTRIDE) + brow * LDSTRIDE + bq * 4) * 4u,
                    Wt + (long)bgr * K + kpos + bq * 4);
    __builtin_prefetch(Ag + (long)(tid >> 2) * K + kpos + 16, 0, 0);
  };

  v8f c00 = {}, c01 = {}, c10 = {}, c11 = {};

  int nk = K >> 4;
  load_tiles(0, 0);
  wait_async0();
  __syncthreads();

  for (int kc = 0; kc < nk; ++kc) {
    int cur = kc & 1;
    if (kc + 1 < nk) load_tiles(cur ^ 1, (kc + 1) << 4);  // overlap with compute

    const float* as = &As[cur][0];
    const float* bs = &Bs[cur][0];
#pragma unroll
    for (int ks = 0; ks < 4; ++ks) {
      int ko = ks * 4 + 2 * half;
      v2f a0 = *(const v2f*)(as + (wm * 32 + row) * LDSTRIDE + ko);
      v2f a1 = *(const v2f*)(as + (wm * 32 + 16 + row) * LDSTRIDE + ko);
      v2f b0 = *(const v2f*)(bs + (wn * 32 + row) * LDSTRIDE + ko);
      v2f b1 = *(const v2f*)(bs + (wn * 32 + 16 + row) * LDSTRIDE + ko);
      c00 = wmma_k4(a0, b0, c00);
      c01 = wmma_k4(a0, b1, c01);
      c10 = wmma_k4(a1, b0, c10);
      c11 = wmma_k4(a1, b1, c11);
    }
    wait_async0();     // next buffer resident (this wave's slice)
    __syncthreads();   // ... and everyone's; current buffer fully consumed
  }

  int cn0 = bn + wn * 32 + row;
  int cn1 = cn0 + 16;
  float bv0 = 0.0f, bv1 = 0.0f;
  if (has_bias) {
    if (cn0 < Ncol) bv0 = bias[cn0];
    if (cn1 < Ncol) bv1 = bias[cn1];
  }
#pragma unroll
  for (int r = 0; r < 8; ++r) {
    long m0 = bm + wm * 32 + r + half * 8;
    long m1 = m0 + 16;
    if (cn0 < Ncol) {
      Cmat[m0 * Ncol + cn0] = c00[r] + bv0;
      Cmat[m1 * Ncol + cn0] = c10[r] + bv0;
    }
    if (cn1 < Ncol) {
      Cmat[m0 * Ncol + cn1] = c01[r] + bv1;
      Cmat[m1 * Ncol + cn1] = c11[r] + bv1;
    }
  }
}

// ---------------------------------------------------------------------------
// BN helpers
// ---------------------------------------------------------------------------
__global__ void zero_kernel(float* __restrict__ p, int n) {
  int i = blockIdx.x * blockDim.x + threadIdx.x;
  if (i < n) p[i] = 0.0f;
}

// gm:[M,C] spatial-major; blockDim.x == C; thread c reduces channel c over rows.
__global__ void bn_stats_kernel(const float* __restrict__ gm, float* __restrict__ stats,
                                int M, int C) {
  int c = threadIdx.x;
  float s = 0.0f, q = 0.0f;
  for (int m = blockIdx.x; m < M; m += gridDim.x) {
    float v = gm[(long)m * C + c];
    s += v;
    q += v * v;
  }
  atomicAdd(&stats[c], s);
  atomicAdd(&stats[C + c], q);
}

// stage-1 tail: BN + ReLU, transpose [M,C] -> channel-major [N,C,S]
__global__ void bn_relu_transpose_kernel(const float* __restrict__ gm,
                                         const float* __restrict__ stats,
                                         const float* __restrict__ gamma,
                                         const float* __restrict__ beta,
                                         float* __restrict__ out, int M, int C, int S) {
  long idx = (long)blockIdx.x * blockDim.x + threadIdx.x;
  long total = (long)M * C;
  if (idx >= total) return;
  int c = (int)(idx % C);
  long m = idx / C;
  float invM = 1.0f / (float)M;
  float mean = stats[c] * invM;
  float var = stats[C + c] * invM - mean * mean;
  float rs = rsqrtf(var + BN_EPS);
  float v = gamma[c] * (gm[idx] - mean) * rs + beta[c];
  v = fmaxf(v, 0.0f);
  long n = m / S, s = m % S;
  out[(n * C + c) * S + s] = v;
}

// stage-2 tail: BN + concat(residual,residual) add + ReLU -> d_out [N,C,S]
__global__ void bn_res_relu_kernel(const float* __restrict__ gm,
                                   const float* __restrict__ stats,
                                   const float* __restrict__ gamma,
                                   const float* __restrict__ beta,
                                   const float* __restrict__ res,  // [N,Cres,S]
                                   float* __restrict__ out, int M, int C, int S, int Cres) {
  long idx = (long)blockIdx.x * blockDim.x + threadIdx.x;
  long total = (long)M * C;
  if (idx >= total) return;
  int c = (int)(idx % C);
  long m = idx / C;
  float invM = 1.0f / (float)M;
  float mean = stats[c] * invM;
  float var = stats[C + c] * invM - mean * mean;
  float rs = rsqrtf(var + BN_EPS);
  float v = gamma[c] * (gm[idx] - mean) * rs + beta[c];
  long n = m / S, s = m % S;
  int rc = c % Cres;
  v += res[(n * Cres + rc) * S + s];
  out[(n * C + c) * S + s] = fmaxf(v, 0.0f);
}

// ---------------------------------------------------------------------------
// launch
// ---------------------------------------------------------------------------
static inline unsigned cdiv(long a, long b) { return (unsigned)((a + b - 1) / b); }

extern "C" void kernel_launch(void* const* d_in, const int* in_sizes, int n_in,
                              void* d_out, int out_size, void* d_ws, size_t ws_size,
                              hipStream_t stream) {
  const float* x        = (const float*)d_in[0];
  const float* off1_w   = (const float*)d_in[1];
  const float* off1_b   = (const float*)d_in[2];
  const float* conv1_w  = (const float*)d_in[3];
  const float* bn1_g    = (const float*)d_in[4];
  const float* bn1_b    = (const float*)d_in[5];
  const float* off2_w   = (const float*)d_in[6];
  const float* off2_b   = (const float*)d_in[7];
  const float* conv2_w  = (const float*)d_in[8];
  const float* bn2_g    = (const float*)d_in[9];
  const float* bn2_b    = (const float*)d_in[10];
  float* out = (float*)d_out;

  const int Nb = 2, Cin = 64, D = 16, H = 56, W = 56, Cout = 128;
  const int Do = 8, Ho = 28, Wo = 28;
  const int S = Do * Ho * Wo;        // 6272
  const long M = (long)Nb * S;       // 12544 = 98 * 128
  const int G1 = Cin / CPG;          // 8  -> OC1 = 648
  const int G2 = Cout / CPG;         // 16 -> OC2 = 1296
  const int OC1 = G1 * 81, OC2 = G2 * 81;
  const int K1 = Cin * 27;           // 1728
  const int K2 = Cout * 27;          // 3456

  // workspace layout (floats)
  float* ws = (float*)d_ws;
  float* res   = ws;                                  // Nb*Cin*S
  float* out1  = res   + (long)Nb * Cin * S;          // Nb*Cout*S (channel-major)
  float* gmout = out1  + (long)Nb * Cout * S;         // M*Cout (spatial-major)
  float* stats = gmout + M * Cout;                    // 2*Cout
  float* offb  = stats + 2 * Cout;                    // M*OC2 max
  float* colsb = offb  + M * OC2;                     // M*K2 max

  const int TB = 256;

  // residual = maxpool3d(x)
  maxpool3d_kernel<<<cdiv((long)Nb * Cin * S, TB), TB, 0, stream>>>(
      x, res, Nb, Cin, D, H, W, Do, Ho, Wo);

  // ---- stage 1 ----
  im2col3d_kernel<<<cdiv((long)Nb * 27 * S, TB), TB, 0, stream>>>(
      x, colsb, Nb, Cin, D, H, W, Do, Ho, Wo, 2);
  {
    dim3 g(cdiv(M, 128), cdiv(OC1, 64));
    gemm_wmma_f32_kernel<<<g, TB, 0, stream>>>(colsb, off1_w, off1_b, offb,
                                               (int)M, K1, OC1, 1);
  }
  deform_cols_kernel<<<cdiv((long)Nb * G1 * 27 * S, TB), TB, 0, stream>>>(
      x, offb, colsb, Nb, Cin, D, H, W, G1, Do, Ho, Wo, 2);
  {
    dim3 g(cdiv(M, 128), cdiv(Cout, 64));
    gemm_wmma_f32_kernel<<<g, TB, 0, stream>>>(colsb, conv1_w, nullptr, gmout,
                                               (int)M, K1, Cout, 0);
  }
  zero_kernel<<<1, 2 * Cout, 0, stream>>>(stats, 2 * Cout);
  bn_stats_kernel<<<256, Cout, 0, stream>>>(gmout, stats, (int)M, Cout);
  bn_relu_transpose_kernel<<<cdiv(M * Cout, TB), TB, 0, stream>>>(
      gmout, stats, bn1_g, bn1_b, out1, (int)M, Cout, S);

  // ---- stage 2 ----
  im2col3d_kernel<<<cdiv((long)Nb * 27 * S, TB), TB, 0, stream>>>(
      out1, colsb, Nb, Cout, Do, Ho, Wo, Do, Ho, Wo, 1);
  {
    dim3 g(cdiv(M, 128), cdiv(OC2, 64));
    gemm_wmma_f32_kernel<<<g, TB, 0, stream>>>(colsb, off2_w, off2_b, offb,
                                               (int)M, K2, OC2, 1);
  }
  deform_cols_kernel<<<cdiv((long)Nb * G2 * 27 * S, TB), TB, 0, stream>>>(
      out1, offb, colsb, Nb, Cout, Do, Ho, Wo, G2, Do, Ho, Wo, 1);
  {
    dim3 g(cdiv(M, 128), cdiv(Cout, 64));
    gemm_wmma_f32_kernel<<<g, TB, 0, stream>>>(colsb, conv2_w, nullptr, gmout,
                                               (int)M, K2, Cout, 0);
  }
  zero_kernel<<<1, 2 * Cout, 0, stream>>>(stats, 2 * Cout);
  bn_stats_kernel<<<256, Cout, 0, stream>>>(gmout, stats, (int)M, Cout);
  bn_res_relu_kernel<<<cdiv(M * Cout, TB), TB, 0, stream>>>(
      gmout, stats, bn2_g, bn2_b, res, out, (int)M, Cout, S, Cin);
}